// Net_40063454937539
// MI455X (gfx1250) — compile-verified
//
#include <hip/hip_runtime.h>
#include <hip/hip_bf16.h>
#include <math.h>

typedef float v2f __attribute__((ext_vector_type(2)));
typedef float v8f __attribute__((ext_vector_type(8)));

// exact pointee type demanded by __builtin_amdgcn_global_load_async_to_lds_b128
typedef int vsi4 __attribute__((__vector_size__(4 * sizeof(int))));
typedef __attribute__((address_space(1))) vsi4 g_vsi4;   // global (AS1)
typedef __attribute__((address_space(3))) vsi4 l_vsi4;   // LDS    (AS3)

#define F_IN 512
#define HID  64
#define NCLS 40
#define NPAD 48

#if defined(__has_builtin)
#if __has_builtin(__builtin_amdgcn_global_load_async_to_lds_b128)
#define HAVE_ASYNC_LDS 1
#endif
#if __has_builtin(__builtin_amdgcn_s_wait_asynccnt)
#define HAVE_WAIT_ASYNC 1
#endif
#endif

__device__ __forceinline__ void wait_asynccnt0() {
#if defined(HAVE_WAIT_ASYNC)
    __builtin_amdgcn_s_wait_asynccnt(0);
#else
    asm volatile("s_wait_asynccnt 0x0" ::: "memory");
#endif
}

// ---------------- GEMM1: h = x @ W1 + b1   [N,512]x[512,64] ----------------
// 1 block = 16 rows, 4 waves; wave w owns output cols [16w,16w+16).
// x tile staged in LDS via async global->LDS DMA (ASYNCcnt path) when the
// toolchain exposes it; row stride 516 -> 4-bank skew, conflict-free ds reads.
__global__ __launch_bounds__(128) void gemm1_wmma(
    const float* __restrict__ x, const float* __restrict__ W1,
    const float* __restrict__ b1, float* __restrict__ h, int N)
{
    constexpr int ROWS = 16;
    constexpr int LDSROW = F_IN + 4;               // 516
    __shared__ __align__(16) float xs[ROWS * LDSROW];

    const int tid = threadIdx.x;
    const int rowBase = blockIdx.x * ROWS;

#if defined(HAVE_ASYNC_LDS)
    for (int i = tid; i < ROWS * F_IN / 4; i += 128) {
        int r = i >> 7;                            // / (512/4)
        int c4 = i & 127;
        int row = rowBase + r;
        float* lp = xs + r * LDSROW + c4 * 4;
        if (row < N) {
            __builtin_amdgcn_global_load_async_to_lds_b128(
                (g_vsi4*)(x + (size_t)row * F_IN + 4 * c4),
                (l_vsi4*)lp, 0, 0);
        } else {
            *(float4*)lp = make_float4(0.f, 0.f, 0.f, 0.f);
        }
    }
    wait_asynccnt0();
#else
    for (int i = tid; i < ROWS * F_IN / 4; i += 128) {
        int r = i >> 7;
        int c4 = i & 127;
        int row = rowBase + r;
        float4 v = make_float4(0.f, 0.f, 0.f, 0.f);
        if (row < N) v = ((const float4*)(x + (size_t)row * F_IN))[c4];
        *(float4*)(xs + r * LDSROW + c4 * 4) = v;
    }
#endif
    __syncthreads();

    const int wave = tid >> 5;
    const int lane = tid & 31;
    const int half = lane >> 4;
    const int ln16 = lane & 15;
    const int colBase = wave * 16;

    v8f c = {};
    const float* arow = xs + ln16 * LDSROW + 2 * half;
    const float* bcol = W1 + (size_t)(2 * half) * HID + colBase + ln16;
    for (int k = 0; k < F_IN; k += 4) {
        v2f a, b;
        a.x = arow[k];
        a.y = arow[k + 1];
        b.x = bcol[(size_t)k * HID];
        b.y = bcol[(size_t)(k + 1) * HID];
        c = __builtin_amdgcn_wmma_f32_16x16x4_f32(false, a, false, b,
                                                  (short)0, c, false, false);
    }

    const float bias = b1[colBase + ln16];
#pragma unroll
    for (int r = 0; r < 8; ++r) {
        int row = rowBase + r + 8 * half;
        if (row < N) h[(size_t)row * HID + colBase + ln16] = c[r] + bias;
    }
}

// ------- GEMM2: h2 = elu(agg1) @ W2 + b2   [N,64]x[64,40] (pad to 48) ------
__global__ __launch_bounds__(96) void gemm2_wmma(
    const float* __restrict__ agg1, const float* __restrict__ W2,
    const float* __restrict__ b2, float* __restrict__ h2, int N)
{
    constexpr int ROWS = 16;
    constexpr int LDSROW = HID + 4;                // 68
    __shared__ float hs[ROWS * LDSROW];
    __shared__ float ws[HID * NPAD];

    const int tid = threadIdx.x;
    const int rowBase = blockIdx.x * ROWS;

    for (int i = tid; i < ROWS * HID; i += 96) {
        int r = i >> 6, cc = i & 63;
        int row = rowBase + r;
        float v = 0.f;
        if (row < N) v = agg1[(size_t)row * HID + cc];
        hs[r * LDSROW + cc] = (v > 0.f) ? v : expm1f(v);   // ELU fused
    }
    for (int i = tid; i < HID * NPAD; i += 96) {
        int r = i / NPAD, cc = i % NPAD;
        ws[i] = (cc < NCLS) ? W2[r * NCLS + cc] : 0.f;
    }
    __syncthreads();

    const int wave = tid >> 5, lane = tid & 31;
    const int half = lane >> 4, ln16 = lane & 15;
    const int colBase = wave * 16;

    v8f c = {};
    const float* arow = hs + ln16 * LDSROW + 2 * half;
    const float* bcol = ws + (2 * half) * NPAD + colBase + ln16;
#pragma unroll
    for (int k = 0; k < HID; k += 4) {
        v2f a, b;
        a.x = arow[k];
        a.y = arow[k + 1];
        b.x = bcol[k * NPAD];
        b.y = bcol[(k + 1) * NPAD];
        c = __builtin_amdgcn_wmma_f32_16x16x4_f32(false, a, false, b,
                                                  (short)0, c, false, false);
    }

    const int col = colBase + ln16;
    const float bias = (col < NCLS) ? b2[col] : 0.f;
#pragma unroll
    for (int r = 0; r < 8; ++r) {
        int row = rowBase + r + 8 * half;
        if (row < N && col < NCLS) h2[(size_t)row * NCLS + col] = c[r] + bias;
    }
}

// -------- edge scatter, 64 channels: one wave32 per edge, 2 atomics/lane ----
__global__ __launch_bounds__(256) void scatter64(
    const long long* __restrict__ src, const long long* __restrict__ dst,
    const float* __restrict__ h, float* __restrict__ agg, int E)
{
    int gw = (blockIdx.x * blockDim.x + threadIdx.x) >> 5;
    int lane = threadIdx.x & 31;
    int nw = (gridDim.x * blockDim.x) >> 5;
    for (int e = gw; e < E; e += nw) {
        long long s = src[e], d = dst[e];
        const float* hp = h + (size_t)s * HID;
        float* ap = agg + (size_t)d * HID;
        unsafeAtomicAdd(&ap[lane], hp[lane]);
        unsafeAtomicAdd(&ap[lane + 32], hp[lane + 32]);
    }
}

// -------- edge scatter, 40 channels -----------------------------------------
__global__ __launch_bounds__(256) void scatter40(
    const long long* __restrict__ src, const long long* __restrict__ dst,
    const float* __restrict__ h, float* __restrict__ agg, int E)
{
    int gw = (blockIdx.x * blockDim.x + threadIdx.x) >> 5;
    int lane = threadIdx.x & 31;
    int nw = (gridDim.x * blockDim.x) >> 5;
    for (int e = gw; e < E; e += nw) {
        long long s = src[e], d = dst[e];
        const float* hp = h + (size_t)s * NCLS;
        float* ap = agg + (size_t)d * NCLS;
        unsafeAtomicAdd(&ap[lane], hp[lane]);
        if (lane < NCLS - 32) unsafeAtomicAdd(&ap[lane + 32], hp[lane + 32]);
    }
}

// -------- log_softmax over 40 classes: one wave32 per row -------------------
__global__ __launch_bounds__(256) void logsoftmax40(
    const float* __restrict__ agg, float* __restrict__ out, int N)
{
    int gw = (blockIdx.x * blockDim.x + threadIdx.x) >> 5;
    int lane = threadIdx.x & 31;
    int nw = (gridDim.x * blockDim.x) >> 5;
    for (int row = gw; row < N; row += nw) {
        const float* p = agg + (size_t)row * NCLS;
        float v0 = p[lane];
        float v1 = (lane < NCLS - 32) ? p[32 + lane] : -INFINITY;
        float m = fmaxf(v0, v1);
        for (int off = 16; off; off >>= 1) m = fmaxf(m, __shfl_xor(m, off, 32));
        float s = expf(v0 - m) + ((lane < NCLS - 32) ? expf(v1 - m) : 0.f);
        for (int off = 16; off; off >>= 1) s += __shfl_xor(s, off, 32);
        float lse = m + logf(s);
        out[(size_t)row * NCLS + lane] = v0 - lse;
        if (lane < NCLS - 32) out[(size_t)row * NCLS + 32 + lane] = v1 - lse;
    }
}

extern "C" void kernel_launch(void* const* d_in, const int* in_sizes, int n_in,
                              void* d_out, int out_size, void* d_ws, size_t ws_size,
                              hipStream_t stream) {
    const float*     x  = (const float*)d_in[0];
    const long long* ei = (const long long*)d_in[1];   // int64 [2,E]
    const float*     W1 = (const float*)d_in[2];
    const float*     b1 = (const float*)d_in[3];
    const float*     W2 = (const float*)d_in[4];
    const float*     b2 = (const float*)d_in[5];
    float* out = (float*)d_out;

    const int N = in_sizes[0] / F_IN;
    const int E = in_sizes[1] / 2;

    float* h    = (float*)d_ws;                 // N*64 (reused as h2: N*40)
    float* agg1 = h    + (size_t)N * HID;       // N*64
    float* agg2 = agg1 + (size_t)N * HID;       // N*40

    (void)hipMemsetAsync(agg1, 0, (size_t)N * HID  * sizeof(float), stream);
    (void)hipMemsetAsync(agg2, 0, (size_t)N * NCLS * sizeof(float), stream);

    const int gblocks = (N + 15) / 16;
    gemm1_wmma<<<gblocks, 128, 0, stream>>>(x, W1, b1, h, N);
    scatter64<<<4096, 256, 0, stream>>>(ei, ei + E, h, agg1, E);
    gemm2_wmma<<<gblocks, 96, 0, stream>>>(agg1, W2, b2, h /* h2 */, N);
    scatter40<<<4096, 256, 0, stream>>>(ei, ei + E, h, agg2, E);
    logsoftmax40<<<1024, 256, 0, stream>>>(agg2, out, N);
}